// MemoryBank_67216238182775
// MI455X (gfx1250) — compile-verified
//
#include <hip/hip_runtime.h>

typedef __attribute__((ext_vector_type(2))) float v2f;
typedef __attribute__((ext_vector_type(8))) float v8f;

#define B_Q    1024
#define D_DIM  256
#define CAPCT  65536
#define TOPK   64

#define ROWS      16            // query rows per workgroup (WMMA M)
#define CHUNK     2048          // columns staged per chunk
#define NCHUNK    (CAPCT / CHUNK)   // 32
#define QS_STRIDE 260           // padded (256 % 64banks == 0 otherwise)
#define SS_STRIDE (CHUNK + 16)  // 2064, padded
#define TL_STRIDE 65            // per-thread top-list stride (bank-conflict-free)

// ---------------------------------------------------------------------------
// usage_new[c] = usage[c] + dot(sum_b Q[b,:], K[c,:])   (deterministic, no atomics)
// ---------------------------------------------------------------------------
__global__ __launch_bounds__(256) void usage_kernel(const float* __restrict__ Q,
                                                    const float* __restrict__ K,
                                                    const float* __restrict__ usage,
                                                    float* __restrict__ out_usage)
{
    __shared__ float qsum[D_DIM];
    const int tid = threadIdx.x;
    float s = 0.0f;
    for (int b = 0; b < B_Q; ++b) s += Q[(size_t)b * D_DIM + tid];
    qsum[tid] = s;
    __syncthreads();

    const int c = blockIdx.x * 256 + tid;
    const float* Krow = K + (size_t)c * D_DIM;
    float acc = usage[c];
    #pragma unroll 8
    for (int k = 0; k < D_DIM; ++k) acc += qsum[k] * Krow[k];
    out_usage[c] = acc;
}

// ---------------------------------------------------------------------------
// Fused: scores tile (WMMA fp32) -> streaming per-row top-64 -> V gather
// One block = 16 query rows, 8 waves. Never materializes scores in HBM.
// ---------------------------------------------------------------------------
__global__ __launch_bounds__(256) void bank_read_kernel(const float* __restrict__ Q,
                                                        const float* __restrict__ K,
                                                        const float* __restrict__ V,
                                                        float* __restrict__ out_h)
{
    __shared__ float Qs[ROWS * QS_STRIDE];     // 16.6 KB  query tile
    __shared__ float Ssc[ROWS * SS_STRIDE];    // 132 KB   score chunk
    __shared__ float topV[256 * TL_STRIDE];    // 66.6 KB  per-thread top-64 values
    __shared__ int   topI[256 * TL_STRIDE];    // 66.6 KB  per-thread top-64 indices
    __shared__ float redV[ROWS * 16];
    __shared__ int   redI[ROWS * 16];
    __shared__ int   winner[ROWS];
    __shared__ int   selIdx[ROWS * TOPK];      // 4 KB     final indices (sorted desc)

    const int tid  = threadIdx.x;
    const int lane = tid & 31;
    const int wave = tid >> 5;     // 0..7
    const int blk  = blockIdx.x;   // 0..63

    // ---- load Q tile (16 x 256) into LDS ----
    for (int i = tid; i < ROWS * D_DIM; i += 256) {
        const int m = i >> 8, k = i & 255;
        Qs[m * QS_STRIDE + k] = Q[(size_t)(blk * ROWS + m) * D_DIM + k];
    }
    // ---- init private top lists ----
    for (int q = 0; q < TOPK; ++q) {
        topV[tid * TL_STRIDE + q] = -__builtin_inff();
        topI[tid * TL_STRIDE + q] = 0x7fffffff;
    }
    float thr = -__builtin_inff();   // current min of my top-64
    int   thrPos = 0;
    __syncthreads();

    const int mRow  = lane & 15;           // WMMA A row / B col within tile
    const int khalf = (lane >> 4) << 1;    // 0 or 2 (K sub-offset per half-wave)
    const int row   = tid >> 4;            // top-k owner: which of 16 rows
    const int sub   = tid & 15;            // top-k owner: column stripe 0..15

    #pragma unroll 1
    for (int chunk = 0; chunk < NCHUNK; ++chunk) {
        const int colBase = chunk * CHUNK;

        // ---- GEMM: wave computes its 256-col stripe as 16 tiles of 16x16 ----
        #pragma unroll 1
        for (int t = 0; t < 16; ++t) {
            const int cLoc = wave * 256 + t * 16 + mRow;   // column within chunk
            const float* Krow = K + (size_t)(colBase + cLoc) * D_DIM;
            v8f acc = {0.f, 0.f, 0.f, 0.f, 0.f, 0.f, 0.f, 0.f};
            #pragma unroll
            for (int kk = 0; kk < D_DIM / 4; ++kk) {       // 64 K-steps of 4
                const int k0 = kk * 4 + khalf;
                v2f a = *(const v2f*)(&Qs[mRow * QS_STRIDE + k0]);
                v2f b = *(const v2f*)(&Krow[k0]);
                acc = __builtin_amdgcn_wmma_f32_16x16x4_f32(
                          false, a, false, b, (short)0, acc, false, false);
            }
            const int mBase = (lane >> 4) * 8;             // C layout: M = r (+8 hi half)
            #pragma unroll
            for (int r = 0; r < 8; ++r)
                Ssc[(mBase + r) * SS_STRIDE + cLoc] = acc[r];
        }
        __syncthreads();

        // ---- streaming top-64: owner (row, sub) scans its 128-col stripe ----
        const int cStart = sub * (CHUNK / 16);
        for (int j = 0; j < CHUNK / 16; ++j) {
            const int c = cStart + j;
            const float v = Ssc[row * SS_STRIDE + c];
            if (v > thr) {
                topV[tid * TL_STRIDE + thrPos] = v;
                topI[tid * TL_STRIDE + thrPos] = colBase + c;
                // recompute running min
                thr = topV[tid * TL_STRIDE + 0]; thrPos = 0;
                for (int q = 1; q < TOPK; ++q) {
                    const float tv = topV[tid * TL_STRIDE + q];
                    if (tv < thr) { thr = tv; thrPos = q; }
                }
            }
        }
        __syncthreads();   // before next chunk overwrites Ssc
    }

    // ---- merge 16 partial lists per row: 64 rounds of argmax extraction ----
    #pragma unroll 1
    for (int round = 0; round < TOPK; ++round) {
        float bv = -__builtin_inff(); int bi = 0x7fffffff, bp = 0;
        for (int q = 0; q < TOPK; ++q) {
            const float v  = topV[tid * TL_STRIDE + q];
            const int   ix = topI[tid * TL_STRIDE + q];
            if (v > bv || (v == bv && ix < bi)) { bv = v; bi = ix; bp = q; }
        }
        redV[row * 16 + sub] = bv;
        redI[row * 16 + sub] = bi;
        __syncthreads();
        if (sub == 0) {
            float gv = -__builtin_inff(); int gi = 0x7fffffff, gw = 0;
            for (int s2 = 0; s2 < 16; ++s2) {
                const float v  = redV[row * 16 + s2];
                const int   ix = redI[row * 16 + s2];
                if (v > gv || (v == gv && ix < gi)) { gv = v; gi = ix; gw = s2; }
            }
            selIdx[row * TOPK + round] = gi;
            winner[row] = gw;
        }
        __syncthreads();
        if (sub == winner[row]) {                     // remove extracted entry
            topV[tid * TL_STRIDE + bp] = -__builtin_inff();
            topI[tid * TL_STRIDE + bp] = 0x7fffffff;
        }
        __syncthreads();
    }

    // ---- gather h_t = V_bank[top_idx] : coalesced float4 copies ----
    const float4* V4   = (const float4*)V;
    float4*       out4 = (float4*)out_h;
    const int D4 = D_DIM / 4;   // 64
    for (int i = tid; i < ROWS * TOPK * D4; i += 256) {
        const int r   = i / (TOPK * D4);
        const int rem = i - r * (TOPK * D4);
        const int j   = rem / D4;
        const int d4  = rem - j * D4;
        const int idx = selIdx[r * TOPK + j];
        out4[((size_t)(blk * ROWS + r) * TOPK + j) * D4 + d4] =
            V4[(size_t)idx * D4 + d4];
    }
}

// ---------------------------------------------------------------------------
extern "C" void kernel_launch(void* const* d_in, const int* in_sizes, int n_in,
                              void* d_out, int out_size, void* d_ws, size_t ws_size,
                              hipStream_t stream)
{
    const float* Q     = (const float*)d_in[0];   // [1024, 256]
    const float* K     = (const float*)d_in[1];   // [65536, 256]
    const float* V     = (const float*)d_in[2];   // [65536, 256]
    const float* usage = (const float*)d_in[3];   // [65536]
    (void)in_sizes; (void)n_in; (void)out_size; (void)d_ws; (void)ws_size;

    float* out_h     = (float*)d_out;                              // [1024,64,256]
    float* out_usage = out_h + (size_t)B_Q * TOPK * D_DIM;         // [65536]

    usage_kernel<<<CAPCT / 256, 256, 0, stream>>>(Q, K, usage, out_usage);
    bank_read_kernel<<<B_Q / ROWS, 256, 0, stream>>>(Q, K, V, out_h);
}